// ContactGraspNetLoss_2508260901650
// MI455X (gfx1250) — compile-verified
//
#include <hip/hip_runtime.h>
#include <math.h>

#define B_  4
#define N_  2048
#define M_  4096
#define BN_ (B_*N_)
#define RADIUS2 (0.005f*0.005f)
#define FARV 100000.0f
#define EPSV 1e-7f

typedef __attribute__((ext_vector_type(2))) float v2f;
typedef __attribute__((ext_vector_type(8))) float v8f;

__constant__ float c_OB[11] = {0.0f, 0.00794435329f, 0.0158887021f, 0.0238330509f,
                               0.0317773996f, 0.0397217484f, 0.0476660972f, 0.0556104459f,
                               0.0635547947f, 0.0714991435f, 0.08f};
__constant__ float c_BW[10] = {0.16652107f, 0.21488856f, 0.37031708f, 0.55618503f,
                               0.75124664f, 0.93943357f, 1.07824539f, 1.19423112f,
                               1.55731375f, 2.34173634f};

// ---------------------------------------------------------------------------
// threefry2x32 replicating jax.random.uniform(jax.random.key(42), (4,2048))
// ---------------------------------------------------------------------------
__device__ __forceinline__ unsigned tf_rotl(unsigned x, unsigned r) {
  return (x << r) | (x >> (32u - r));
}
__device__ __forceinline__ float jax_uniform_threefry(unsigned e) {
  const unsigned ks0 = 0u, ks1 = 42u, ks2 = 0u ^ 42u ^ 0x1BD11BDAu;
  unsigned x0 = (e < 4096u) ? e : (e - 4096u);
  unsigned x1 = (e < 4096u) ? (e + 4096u) : e;
  x0 += ks0; x1 += ks1;
#define TFR(r) { x0 += x1; x1 = tf_rotl(x1, r); x1 ^= x0; }
  TFR(13u) TFR(15u) TFR(26u) TFR(6u)   x0 += ks1; x1 += ks2 + 1u;
  TFR(17u) TFR(29u) TFR(16u) TFR(24u)  x0 += ks2; x1 += ks0 + 2u;
  TFR(13u) TFR(15u) TFR(26u) TFR(6u)   x0 += ks0; x1 += ks1 + 3u;
  TFR(17u) TFR(29u) TFR(16u) TFR(24u)  x0 += ks1; x1 += ks2 + 4u;
  TFR(13u) TFR(15u) TFR(26u) TFR(6u)   x0 += ks2; x1 += ks0 + 5u;
#undef TFR
  unsigned bits = (e < 4096u) ? x0 : x1;
  unsigned f = (bits >> 9) | 0x3F800000u;
  float u = __uint_as_float(f) - 1.0f;
  return fmaxf(u, 0.0f);
}

// ---------------------------------------------------------------------------
// K0: nearest-contact search via V_WMMA_F32_16X16X4_F32 with the norm folded
// into the K=4 pad slot:
//   A row (contact m) = (qx, qy, qz, qn)      [rows of D]
//   B col (pred j)    = (-2px, -2py, -2pz, 1) [cols of D]
//   => acc[r] = qn_m - 2 q_m . p_j ;  d2 = acc[r] + pn_j (lane-local constant)
// No cross-lane shuffles in the hot loop; next tile is prefetched during the
// WMMA + argmin epilogue.
// ---------------------------------------------------------------------------
__global__ __launch_bounds__(32) void nn_kernel(const float* __restrict__ pred_points,
                                                const float* __restrict__ contact_points,
                                                int* __restrict__ idx_out,
                                                float* __restrict__ d2_out) {
  const int tile = blockIdx.x;                 // B_*N_/16 = 512 tiles
  const int b  = tile / (N_ / 16);
  const int n0 = (tile % (N_ / 16)) * 16;
  const int lane = threadIdx.x;
  const int li = lane & 15, hf = lane >> 4;

  const float* pp = pred_points + ((size_t)b * N_ + n0 + li) * 3;
  const float px = pp[0], py = pp[1], pz = pp[2];
  const float pn = px * px + py * py + pz * pz;
  v2f bfrag;
  bfrag.x = hf ? (-2.0f * pz) : (-2.0f * px);
  bfrag.y = hf ? 1.0f : (-2.0f * py);

  float best = 3.4e38f;
  int bestIdx = 0;

  const float* qbase = contact_points + (size_t)b * M_ * 3;
  float qx, qy, qz;
  {
    const float* qp = qbase + (size_t)li * 3;
    qx = qp[0]; qy = qp[1]; qz = qp[2];
  }
  for (int m0 = 0; m0 < M_; m0 += 16) {
    // prefetch next contact tile
    float nqx = 0.f, nqy = 0.f, nqz = 0.f;
    if (m0 + 16 < M_) {
      const float* qp = qbase + (size_t)(m0 + 16 + li) * 3;
      nqx = qp[0]; nqy = qp[1]; nqz = qp[2];
    }
    const float qn = qx * qx + qy * qy + qz * qz;
    v2f afrag;
    afrag.x = hf ? qz : qx;
    afrag.y = hf ? qn : qy;

    v8f acc = {0.f, 0.f, 0.f, 0.f, 0.f, 0.f, 0.f, 0.f};
    acc = __builtin_amdgcn_wmma_f32_16x16x4_f32(false, afrag, false, bfrag,
                                                (short)0, acc, false, false);
    const int rowbase = m0 + hf * 8;
#pragma unroll
    for (int r = 0; r < 8; ++r) {
      const float d2 = acc[r] + pn;
      const int mi = rowbase + r;
      const bool better = d2 < best;
      best = better ? d2 : best;
      bestIdx = better ? mi : bestIdx;
    }
    qx = nqx; qy = nqy; qz = nqz;
  }
  const float ob = __shfl(best, lane + 16, 32);
  const int oi = __shfl(bestIdx, lane + 16, 32);
  if (lane < 16) {
    if (ob < best) { best = ob; bestIdx = oi; }
    const int id = b * N_ + n0 + li;
    idx_out[id] = bestIdx;
    d2_out[id] = best;
  }
}

// ---------------------------------------------------------------------------
// K1: per-point prep: pos/bce/width loss, threefry priorities, build padded
// 16-float control-point vectors:
//   Pm2 = -2 * P with slot15 = 1.0  (B-side of ADDS WMMA)
//   G   =  G     with slot15 = |G|^2, Gs likewise (A-side, norm folded in K)
// batch_acc[b*8 + {0:npos, 1:wlpos, 2:bce_sel, 3:sel, 4:adds}]
// ---------------------------------------------------------------------------
__global__ void prep_kernel(const float* __restrict__ pred_grasps,
                            const float* __restrict__ pred_scores,
                            const float* __restrict__ gwh,
                            const float* __restrict__ pcw,
                            const float* __restrict__ pcr,
                            const float* __restrict__ pct,
                            const float* __restrict__ cp,
                            const float* __restrict__ cps,
                            const int* __restrict__ idx_in,
                            const float* __restrict__ d2_in,
                            float* __restrict__ pos_out,
                            float* __restrict__ bce_out,
                            float* __restrict__ pri_out,
                            float* __restrict__ Pm2,
                            float* __restrict__ G16,
                            float* __restrict__ Gs16,
                            float* __restrict__ pn_out,
                            float* __restrict__ batch_acc) {
  const int id = blockIdx.x * blockDim.x + threadIdx.x;
  if (id >= BN_) return;
  const int b = id / N_, n = id % N_;

  const int ci = idx_in[id];
  const float md2 = d2_in[id];
  const float pos = (md2 < RADIUS2) ? 1.0f : 0.0f;
  pos_out[id] = pos;
  atomicAdd(&batch_acc[b * 8 + 0], pos);

  float p = pred_scores[id];
  p = fminf(fmaxf(p, EPSV), 1.0f - EPSV);
  bce_out[id] = -(pos * logf(p) + (1.0f - pos) * logf(1.0f - p));

  // width multihot loss (mean over 10 weighted bins)
  const float w = pcw[b * M_ + ci];
  float wl = 0.0f;
#pragma unroll
  for (int k = 0; k < 10; ++k) {
    const float mh = (w >= c_OB[k] && w < c_OB[k + 1]) ? 1.0f : 0.0f;
    const float x = gwh[((size_t)b * 10 + k) * N_ + n];
    const float bw = fmaxf(x, 0.0f) - x * mh + log1pf(expf(-fabsf(x)));
    wl += c_BW[k] * bw;
  }
  wl *= 0.1f;
  atomicAdd(&batch_acc[b * 8 + 1], wl * pos);

  // negative-mining priority
  const float u = jax_uniform_threefry((unsigned)id);
  pri_out[id] = (pos > 0.5f) ? __builtin_inff() : u;

  // rot/trans labels (FAR where not a positive)
  float Rl[9], tl[3];
  if (pos > 0.5f) {
    const float* rp = pcr + (size_t)(b * M_ + ci) * 9;
#pragma unroll
    for (int q = 0; q < 9; ++q) Rl[q] = rp[q];
    const float* tp = pct + (size_t)(b * M_ + ci) * 3;
    tl[0] = tp[0]; tl[1] = tp[1]; tl[2] = tp[2];
  } else {
#pragma unroll
    for (int q = 0; q < 9; ++q) Rl[q] = FARV;
    tl[0] = FARV; tl[1] = FARV; tl[2] = FARV;
  }

  // predicted rot/trans
  const float* pg = pred_grasps + (size_t)id * 16;
  float PR[9], pt[3];
#pragma unroll
  for (int i = 0; i < 3; ++i) {
#pragma unroll
    for (int j = 0; j < 3; ++j) PR[i * 3 + j] = pg[i * 4 + j];
    pt[i] = pg[i * 4 + 3];
  }

  float pnv = 0.f, gnv = 0.f, gsnv = 0.f;
  const size_t base = (size_t)id * 16;
#pragma unroll
  for (int c = 0; c < 5; ++c) {
    const float cx = cp[c * 3], cy = cp[c * 3 + 1], cz = cp[c * 3 + 2];
    const float sx = cps[c * 3], sy = cps[c * 3 + 1], sz = cps[c * 3 + 2];
#pragma unroll
    for (int i = 0; i < 3; ++i) {
      const float Pv = PR[i * 3] * cx + PR[i * 3 + 1] * cy + PR[i * 3 + 2] * cz + pt[i];
      const float Gv = Rl[i * 3] * cx + Rl[i * 3 + 1] * cy + Rl[i * 3 + 2] * cz + tl[i];
      const float Gs = Rl[i * 3] * sx + Rl[i * 3 + 1] * sy + Rl[i * 3 + 2] * sz + tl[i];
      Pm2[base + c * 3 + i] = -2.0f * Pv;
      G16[base + c * 3 + i] = Gv;
      Gs16[base + c * 3 + i] = Gs;
      pnv += Pv * Pv; gnv += Gv * Gv; gsnv += Gs * Gs;
    }
  }
  Pm2[base + 15] = 1.0f;        // multiplies the norm slot of G/Gs
  G16[base + 15] = gnv;         // |G|^2 folded into the K reduction
  Gs16[base + 15] = gsnv;
  pn_out[id] = pnv;
}

// ---------------------------------------------------------------------------
// K2: hard-negative selection: sel_neg = (rank of pri among negatives) < k
// via brute-force count (positives hold +inf priority, never counted).
// ---------------------------------------------------------------------------
__global__ void sel_kernel(const float* __restrict__ pos_in,
                           const float* __restrict__ pri_in,
                           const float* __restrict__ bce_in,
                           float* __restrict__ batch_acc) {
  const int id = blockIdx.x * blockDim.x + threadIdx.x;
  if (id >= BN_) return;
  const int b = id / N_;
  const float npos = batch_acc[b * 8 + 0];
  const float k = (npos > 0.0f) ? npos : 2.0f;

  float sel;
  if (pos_in[id] > 0.5f) {
    sel = 1.0f;
  } else {
    const float pri_i = pri_in[id];
    const float* pb = pri_in + (size_t)b * N_;
    int cnt = 0;
    for (int j = 0; j < N_; ++j) cnt += (pb[j] < pri_i) ? 1 : 0;
    sel = ((float)cnt < k) ? 1.0f : 0.0f;
  }
  atomicAdd(&batch_acc[b * 8 + 2], bce_in[id] * sel);
  atomicAdd(&batch_acc[b * 8 + 3], sel);
}

// ---------------------------------------------------------------------------
// K3: ADDS min-distance via chained V_WMMA_F32_16X16X4_F32 (K = 16 = 15+norm).
// One wave per (b, 16-col tile). A = label rows (G/Gs with |.|^2 in slot 15),
// B = -2*P cols with 1.0 in slot 15; so after the 4-WMMA chain
//   acc[r] = |G_row|^2 - 2<G_row, P_col>  and  d2 = acc[r] + pn_col.
// Min over rows is pure in-lane VALU; next j-tile A-data is prefetched.
// ---------------------------------------------------------------------------
__global__ __launch_bounds__(32) void adds_kernel(const float* __restrict__ Pm2,
                                                  const float* __restrict__ G16,
                                                  const float* __restrict__ Gs16,
                                                  const float* __restrict__ pn_in,
                                                  const float* __restrict__ pos_in,
                                                  const float* __restrict__ pred_scores,
                                                  float* __restrict__ batch_acc) {
  const int tile = blockIdx.x;                 // 512 tiles
  const int b  = tile / (N_ / 16);
  const int i0 = (tile % (N_ / 16)) * 16;
  const int lane = threadIdx.x;
  const int li = lane & 15, hf = lane >> 4;

  const int colId = b * N_ + i0 + li;
  const float* Prow = Pm2 + (size_t)colId * 16;
  v2f bfr[4];
#pragma unroll
  for (int kk = 0; kk < 4; ++kk)
    bfr[kk] = *(const v2f*)(Prow + 4 * kk + 2 * hf);
  const float pn = pn_in[colId];

  float best = 3.4e38f;
  const size_t rowBase0 = ((size_t)b * N_ + li) * 16 + 2 * hf;

  v2f aG[4], aS[4];
#pragma unroll
  for (int kk = 0; kk < 4; ++kk) {
    aG[kk] = *(const v2f*)(G16 + rowBase0 + 4 * kk);
    aS[kk] = *(const v2f*)(Gs16 + rowBase0 + 4 * kk);
  }

  for (int j0 = 0; j0 < N_; j0 += 16) {
    // prefetch next j-tile rows for both variants
    v2f nG[4] = {}, nS[4] = {};
    if (j0 + 16 < N_) {
      const size_t rb = rowBase0 + (size_t)(j0 + 16) * 16;
#pragma unroll
      for (int kk = 0; kk < 4; ++kk) {
        nG[kk] = *(const v2f*)(G16 + rb + 4 * kk);
        nS[kk] = *(const v2f*)(Gs16 + rb + 4 * kk);
      }
    }
    // variant 1: G
    {
      v8f acc = {0.f, 0.f, 0.f, 0.f, 0.f, 0.f, 0.f, 0.f};
#pragma unroll
      for (int kk = 0; kk < 4; ++kk)
        acc = __builtin_amdgcn_wmma_f32_16x16x4_f32(false, aG[kk], false, bfr[kk],
                                                    (short)0, acc, false, false);
#pragma unroll
      for (int r = 0; r < 8; ++r) best = fminf(best, acc[r] + pn);
    }
    // variant 2: Gs (symmetric gripper)
    {
      v8f acc = {0.f, 0.f, 0.f, 0.f, 0.f, 0.f, 0.f, 0.f};
#pragma unroll
      for (int kk = 0; kk < 4; ++kk)
        acc = __builtin_amdgcn_wmma_f32_16x16x4_f32(false, aS[kk], false, bfr[kk],
                                                    (short)0, acc, false, false);
#pragma unroll
      for (int r = 0; r < 8; ++r) best = fminf(best, acc[r] + pn);
    }
#pragma unroll
    for (int kk = 0; kk < 4; ++kk) { aG[kk] = nG[kk]; aS[kk] = nS[kk]; }
  }
  const float ob = __shfl(best, lane + 16, 32);
  if (lane < 16) {
    best = fminf(best, ob);
    const float m = fmaxf(best, 0.0f);
    const float mad = pos_in[colId] * sqrtf(m + 1e-12f);
    atomicAdd(&batch_acc[b * 8 + 4], pred_scores[colId] * mad);
  }
}

// ---------------------------------------------------------------------------
// K4: finalize the 4 scalar outputs.
// ---------------------------------------------------------------------------
__global__ void finalize_kernel(const float* __restrict__ batch_acc,
                                float* __restrict__ out) {
  if (blockIdx.x != 0 || threadIdx.x != 0) return;
  float bc = 0.f, wd = 0.f, ad = 0.f;
#pragma unroll
  for (int b = 0; b < B_; ++b) {
    const float npos = batch_acc[b * 8 + 0];
    const float piv = fmaxf(npos, 1.0f);
    bc += batch_acc[b * 8 + 2] / fmaxf(batch_acc[b * 8 + 3], 1.0f);
    wd += batch_acc[b * 8 + 1] / piv;
    ad += batch_acc[b * 8 + 4] / piv;
  }
  bc *= (1.0f / B_); wd *= (1.0f / B_); ad *= (1.0f / B_);
  out[0] = bc + wd + 3.0f * ad;
  out[1] = bc;
  out[2] = wd;
  out[3] = ad;
}

__global__ void init_kernel(float* __restrict__ batch_acc) {
  const int i = blockIdx.x * blockDim.x + threadIdx.x;
  if (i < 8 * B_) batch_acc[i] = 0.0f;
}

// ---------------------------------------------------------------------------
extern "C" void kernel_launch(void* const* d_in, const int* in_sizes, int n_in,
                              void* d_out, int out_size, void* d_ws, size_t ws_size,
                              hipStream_t stream) {
  const float* pred_grasps = (const float*)d_in[0];
  const float* pred_scores = (const float*)d_in[1];
  const float* pred_points = (const float*)d_in[2];
  const float* gwh         = (const float*)d_in[3];
  const float* pcp         = (const float*)d_in[4];
  const float* pcw         = (const float*)d_in[5];
  const float* pcr         = (const float*)d_in[6];
  const float* pct         = (const float*)d_in[7];
  const float* gcp         = (const float*)d_in[8];
  const float* gcps        = (const float*)d_in[9];
  float* out = (float*)d_out;

  float* ws = (float*)d_ws;
  int*   idx  = (int*)ws;             // BN_
  float* d2   = ws + 1 * BN_;         // BN_
  float* pos  = ws + 2 * BN_;
  float* bce  = ws + 3 * BN_;
  float* pri  = ws + 4 * BN_;
  float* pn   = ws + 5 * BN_;
  float* Pm2  = ws + 6 * BN_;         // BN_*16
  float* G16  = ws + 22 * BN_;        // BN_*16
  float* Gs16 = ws + 38 * BN_;        // BN_*16
  float* bacc = ws + 54 * BN_;        // 32 floats

  init_kernel<<<1, 64, 0, stream>>>(bacc);
  nn_kernel<<<BN_ / 16, 32, 0, stream>>>(pred_points, pcp, idx, d2);
  prep_kernel<<<BN_ / 256, 256, 0, stream>>>(pred_grasps, pred_scores, gwh, pcw, pcr, pct,
                                             gcp, gcps, idx, d2, pos, bce, pri,
                                             Pm2, G16, Gs16, pn, bacc);
  sel_kernel<<<BN_ / 256, 256, 0, stream>>>(pos, pri, bce, bacc);
  adds_kernel<<<BN_ / 16, 32, 0, stream>>>(Pm2, G16, Gs16, pn, pos,
                                           pred_scores, bacc);
  finalize_kernel<<<1, 32, 0, stream>>>(bacc, out);
}